// WeightedHausdorffDistance_21328807592099
// MI455X (gfx1250) — compile-verified
//
#include <hip/hip_runtime.h>
#include <hip/hip_bf16.h>

typedef float v2f __attribute__((ext_vector_type(2)));
typedef float v8f __attribute__((ext_vector_type(8)));

#define HH 256
#define WW 256
#define BB 8
#define MM 256
#define NN (HH * WW)

__device__ __forceinline__ float f_inf() { return __uint_as_float(0x7F800000u); }

static constexpr float EPSF  = 1e-6f;
static constexpr float MAXD  = 362.03867196751236f; // sqrt(256^2 + 256^2)

// ---------------------------------------------------------------------------
// Kernel 0: initialize workspace (col-min bits = +inf, per-batch sums = 0)
// ---------------------------------------------------------------------------
__global__ void whd_init(unsigned* __restrict__ colBits, float* __restrict__ sums) {
    int idx = blockIdx.x * 256 + threadIdx.x;
    if (idx < BB * MM) colBits[idx] = 0x7F800000u;
    if (idx < 2 * BB)  sums[idx] = 0.0f;
}

// ---------------------------------------------------------------------------
// Kernel 1: main WMMA pass.
// grid = BB*512 blocks, 256 threads (8 waves). Wave w of block rb handles
// pixel rows [rb*128 + w*16, +16) of batch b against all 256 GT points,
// 16 columns per WMMA tile.
//
// d2[i][j] = [nx0, nx1, |nx|^2, 1] . [-2ny0, -2ny1, 1, |ny|^2]  (K=4, exact)
// ---------------------------------------------------------------------------
__global__ __launch_bounds__(256) void whd_main(const float* __restrict__ prob,
                                                const float* __restrict__ gt,
                                                const float* __restrict__ osz,
                                                unsigned* __restrict__ colBits,
                                                float* __restrict__ sums) {
    __shared__ unsigned ldsCol[MM];

    const int tid  = threadIdx.x;
    const int b    = blockIdx.x >> 9;   // 512 blocks per batch
    const int rb   = blockIdx.x & 511;
    const int wv   = tid >> 5;
    const int lane = tid & 31;
    const int n    = lane & 15;         // A-row index / B-column index for this lane
    const int sel  = lane >> 4;         // K-pair selector (layout half)

    for (int j = tid; j < MM; j += 256) ldsCol[j] = 0x7F800000u;
    __syncthreads();

    const float nf0 = osz[2 * b + 0] * (1.0f / HH);
    const float nf1 = osz[2 * b + 1] * (1.0f / WW);

    // ---- A fragment: 16x4 f32, lanes 0-15 hold K={0,1}, lanes 16-31 K={2,3}
    const int rowBase = rb * 128 + wv * 16;
    const int i  = rowBase + n;                 // flat pixel id of row n
    const float py  = (float)(i >> 8);
    const float px  = (float)(i & 255);
    const float nx0 = py * nf0, nx1 = px * nf1;
    const float nxsq = nx0 * nx0 + nx1 * nx1;
    v2f A;
    A.x = sel ? nxsq : nx0;
    A.y = sel ? 1.0f : nx1;

    // p and 1/(p^4 + eps/maxd) for this lane's row n (duplicated across halves)
    const float p    = prob[b * NN + i];
    const float p2   = p * p;
    const float invw = 1.0f / (p2 * p2 + EPSF / MAXD);

    // Hoisted: invw for row v+8*sel (loop-invariant — 8 bpermutes ONCE, not per tile)
    float invwv[8];
#pragma unroll
    for (int v = 0; v < 8; ++v) invwv[v] = __shfl(invw, v + 8 * sel, 16);

    float rmin[8];
#pragma unroll
    for (int v = 0; v < 8; ++v) rmin[v] = f_inf();

    const float* gtb = gt + b * MM * 2;

#pragma unroll 1
    for (int t = 0; t < 16; ++t) {
        const int j = t * 16 + n;               // global GT column
        const float gy  = gtb[2 * j + 0];
        const float gx  = gtb[2 * j + 1];
        const float ny0 = gy * nf0, ny1 = gx * nf1;
        v2f Bf;
        if (sel) { Bf.x = 1.0f;        Bf.y = ny0 * ny0 + ny1 * ny1; }
        else     { Bf.x = -2.0f * ny0; Bf.y = -2.0f * ny1;           }

        v8f c = {0.f, 0.f, 0.f, 0.f, 0.f, 0.f, 0.f, 0.f};
        // v_wmma_f32_16x16x4_f32: full 16x16 tile of squared distances
        c = __builtin_amdgcn_wmma_f32_16x16x4_f32(false, A, false, Bf,
                                                  (short)0, c, false, false);

        // C/D layout: VGPR v, lanes 0-15 -> row v, lanes 16-31 -> row v+8; col = n
        float scmin = f_inf();
#pragma unroll
        for (int v = 0; v < 8; ++v) {
            // fast hardware sqrt: single v_sqrt_f32 (matches GPU reference tolerance)
            float d = __builtin_amdgcn_sqrtf(__builtin_fmaxf(c[v], 0.0f));
            rmin[v] = __builtin_fminf(rmin[v], d);
            scmin = __builtin_fminf(scmin, (d + EPSF) * invwv[v]);
        }
        // combine the two halves (same column, rows 0-7 vs 8-15)
        scmin = __builtin_fminf(scmin, __shfl_xor(scmin, 16, 32));
        if (lane < 16) atomicMin(&ldsCol[j], __float_as_uint(scmin));
    }

    // ---- row-min over all 256 columns: reduce across the 16 lanes of each half
    float acc = 0.0f;
#pragma unroll
    for (int v = 0; v < 8; ++v) {
        float r = rmin[v];
        r = fminf(r, __shfl_xor(r, 1, 16));
        r = fminf(r, __shfl_xor(r, 2, 16));
        r = fminf(r, __shfl_xor(r, 4, 16));
        r = fminf(r, __shfl_xor(r, 8, 16));      // all 16 lanes now hold rowmin
        float pr = __shfl(p, v + 8 * sel, 16);   // p of row v+8*sel
        acc += pr * r;                           // replicated within half: 8 rows/half
    }
    acc += __shfl_xor(acc, 16, 32);              // wave total: sum p_i * rowmin_i (16 rows)

    float ps = p;                                // sum of p over the wave's 16 rows
    ps += __shfl_xor(ps, 1, 16);
    ps += __shfl_xor(ps, 2, 16);
    ps += __shfl_xor(ps, 4, 16);
    ps += __shfl_xor(ps, 8, 16);

    if (lane == 0) {
        atomicAdd(&sums[b], ps);                 // Sigma p
        atomicAdd(&sums[BB + b], acc);           // Sigma p * rowmin
    }

    __syncthreads();
    for (int j = tid; j < MM; j += 256)
        atomicMin(&colBits[b * MM + j], ldsCol[j]);
}

// ---------------------------------------------------------------------------
// Kernel 2: fold per-batch results into the scalar output.
// out = mean_b( sumPmin_b / (sumP_b + eps) ) + mean_{b,j}( clip(colmin, 0, MAXD) )
// ---------------------------------------------------------------------------
__global__ __launch_bounds__(256) void whd_final(const unsigned* __restrict__ colBits,
                                                 const float* __restrict__ sums,
                                                 float* __restrict__ out) {
    __shared__ float red[256];
    const int j = threadIdx.x;
    float s = 0.0f;
    for (int b = 0; b < BB; ++b) {
        float v = __uint_as_float(colBits[b * MM + j]);
        v = fminf(fmaxf(v, 0.0f), MAXD);
        s += v;
    }
    red[j] = s;
    __syncthreads();
    for (int off = 128; off > 0; off >>= 1) {
        if (j < off) red[j] += red[j + off];
        __syncthreads();
    }
    if (j == 0) {
        float t1 = 0.0f;
        for (int b = 0; b < BB; ++b) t1 += sums[BB + b] / (sums[b] + EPSF);
        out[0] = t1 * (1.0f / BB) + red[0] * (1.0f / (BB * MM));
    }
}

// ---------------------------------------------------------------------------
extern "C" void kernel_launch(void* const* d_in, const int* in_sizes, int n_in,
                              void* d_out, int out_size, void* d_ws, size_t ws_size,
                              hipStream_t stream) {
    const float* prob = (const float*)d_in[0];   // [B, H, W]
    const float* gt   = (const float*)d_in[1];   // [B, M, 2]
    const float* osz  = (const float*)d_in[2];   // [B, 2]

    unsigned* colBits = (unsigned*)d_ws;               // B*M uints
    float*    sums    = (float*)d_ws + BB * MM;        // [0..B): sumP, [B..2B): sum p*rowmin
    float*    out     = (float*)d_out;

    whd_init <<<8, 256, 0, stream>>>(colBits, sums);
    whd_main <<<BB * 512, 256, 0, stream>>>(prob, gt, osz, colBits, sums);
    whd_final<<<1, 256, 0, stream>>>(colBits, sums, out);
}